// STMODEL_13554916786841
// MI455X (gfx1250) — compile-verified
//
#include <hip/hip_runtime.h>
#include <math.h>

#define N_NODES 100000
#define E_EDGES 1600000
#define IN_D    256
#define HID     64
#define OUT_D   32
#define NEG_SLOPE 0.2f
#define EPS_F 1e-16f

typedef __attribute__((ext_vector_type(2))) float v2f;
typedef __attribute__((ext_vector_type(8))) float v8f;

// D = A(16x4 f32) * B(4x16 f32) + C(16x16 f32)
__device__ __forceinline__ v8f wmma_k4(v2f a, v2f b, v8f c) {
    return __builtin_amdgcn_wmma_f32_16x16x4_f32(false, a, false, b, (short)0, c, false, false);
}

// sign-aware float atomic max via integer atomics (monotone in both encodings)
__device__ __forceinline__ void atomicMaxF(float* addr, float val) {
    if (val >= 0.0f) atomicMax((int*)addr, __float_as_int(val));
    else             atomicMin((unsigned int*)addr, __float_as_uint(val));
}

// ---------------- GEMM1: H = X[N,256] @ W1[256,64], one 16x16 tile per wave ----
__global__ __launch_bounds__(256) void k_gemm1(const float* __restrict__ X,
                                               const float* __restrict__ W,
                                               float* __restrict__ H) {
    const int wave = threadIdx.x >> 5;          // 8 waves / block
    const int lane = threadIdx.x & 31;
    const int half = lane >> 4, l16 = lane & 15;
    const int rowTile = blockIdx.x * 2 + (wave >> 2);   // 2 row tiles per block
    const int colTile = wave & 3;                        // HID/16 = 4 col tiles
    const int r0 = rowTile * 16;
    const int n0 = colTile * 16;

    v8f acc = {};
    const float* a_base = X + (size_t)(r0 + l16) * IN_D + 2 * half;
    const float* b_base = W + (size_t)(2 * half) * HID + n0 + l16;
    for (int k0 = 0; k0 < IN_D; k0 += 4) {
        v2f a, b;
        a.x = a_base[k0];
        a.y = a_base[k0 + 1];
        b.x = b_base[(size_t)k0 * HID];
        b.y = b_base[(size_t)(k0 + 1) * HID];
        acc = wmma_k4(a, b, acc);
    }
    float* out = H + (size_t)r0 * HID + n0;
#pragma unroll
    for (int r = 0; r < 8; ++r)
        out[(size_t)(r + 8 * half) * HID + l16] = acc[r];
}

// ---------------- per-node attention logits --------------------------------
__global__ void k_alpha(const float* __restrict__ H, const float* __restrict__ a_s,
                        const float* __restrict__ a_d, float* __restrict__ alpha_s,
                        float* __restrict__ alpha_d) {
    int i = blockIdx.x * blockDim.x + threadIdx.x;
    if (i >= N_NODES) return;
    const float* h = H + (size_t)i * HID;
    float s = 0.f, d = 0.f;
#pragma unroll 4
    for (int k = 0; k < HID; ++k) { float v = h[k]; s += v * a_s[k]; d += v * a_d[k]; }
    alpha_s[i] = s;
    alpha_d[i] = d;
}

// ---------------- init scratch (must run every launch) ---------------------
__global__ void k_init(float* __restrict__ h1agg, float* __restrict__ nmax,
                       float* __restrict__ nsum) {
    int i = blockIdx.x * blockDim.x + threadIdx.x;
    if (i < N_NODES * HID) h1agg[i] = 0.0f;
    if (i < N_NODES) { nmax[i] = -__builtin_inff(); nsum[i] = 0.0f; }
}

// ---------------- edge pass 1: leakyrelu + segment max ---------------------
__global__ void k_edge1(const int* __restrict__ src, const int* __restrict__ dst,
                        const float* __restrict__ alpha_s, const float* __restrict__ alpha_d,
                        float* __restrict__ eval_, float* __restrict__ nmax) {
    int e = blockIdx.x * blockDim.x + threadIdx.x;
    if (e >= E_EDGES) return;
    int s = src[e], d = dst[e];
    float v = alpha_s[s] + alpha_d[d];
    v = v > 0.0f ? v : NEG_SLOPE * v;
    eval_[e] = v;
    atomicMaxF(&nmax[d], v);
}

// ---------------- edge pass 2: exp + segment sum ---------------------------
__global__ void k_edge2(const int* __restrict__ dst, float* __restrict__ eval_,
                        const float* __restrict__ nmax, float* __restrict__ nsum) {
    int e = blockIdx.x * blockDim.x + threadIdx.x;
    if (e >= E_EDGES) return;
    int d = dst[e];
    float ex = expf(eval_[e] - nmax[d]);
    eval_[e] = ex;
    atomicAdd(&nsum[d], ex);
}

// ---------------- edge pass 3: weighted scatter of h[src] ------------------
// one wave per edge; each lane handles 2 consecutive columns of HID=64
__global__ void k_edge3(const int* __restrict__ src, const int* __restrict__ dst,
                        const float* __restrict__ eval_, const float* __restrict__ nsum,
                        const float* __restrict__ H, float* __restrict__ h1agg) {
    long long gid = (long long)blockIdx.x * blockDim.x + threadIdx.x;
    int e = (int)(gid >> 5);
    if (e >= E_EDGES) return;
    int lane = (int)(gid & 31);
    int s = src[e], d = dst[e];
    float w = eval_[e] / (nsum[d] + EPS_F);
    int c = lane * 2;
    const float* hp = H + (size_t)s * HID + c;
    float h0 = hp[0], h1 = hp[1];
    float* op = h1agg + (size_t)d * HID + c;
    atomicAdd(op + 0, w * h0);
    atomicAdd(op + 1, w * h1);
}

// ---------------- ELU elementwise ------------------------------------------
__global__ void k_elu(float* __restrict__ x) {
    int i = blockIdx.x * blockDim.x + threadIdx.x;
    if (i >= N_NODES * HID) return;
    float v = x[i];
    x[i] = v > 0.0f ? v : (expf(v) - 1.0f);
}

// ---------------- fused tail: h2 = h1@w2 ; h3 = elu(h2@l1wT+b) ; h4 = h3@l2wT+b
// one 16-node tile per wave; per-wave LDS stage re-layouts C-frags -> A-frags
__global__ __launch_bounds__(256) void k_post(const float* __restrict__ H1,
                                              const float* __restrict__ w2,   // [64,32]
                                              const float* __restrict__ l1w,  // [64,32]
                                              const float* __restrict__ l1b,  // [64]
                                              const float* __restrict__ l2w,  // [256,64]
                                              const float* __restrict__ l2b,  // [256]
                                              float* __restrict__ out_h2,     // [N,32]
                                              float* __restrict__ out_h4) {   // [N,256]
    __shared__ float s_w2[HID * OUT_D];
    __shared__ float s_l1w[HID * OUT_D];
    __shared__ float s_l1b[HID];
    __shared__ float s_l2b[IN_D];
    __shared__ float s_stage[8][16 * HID];   // per-wave 16x64 staging tile

    for (int i = threadIdx.x; i < HID * OUT_D; i += 256) { s_w2[i] = w2[i]; s_l1w[i] = l1w[i]; }
    for (int i = threadIdx.x; i < HID; i += 256) s_l1b[i] = l1b[i];
    for (int i = threadIdx.x; i < IN_D; i += 256) s_l2b[i] = l2b[i];
    __syncthreads();

    const int wave = threadIdx.x >> 5;
    const int lane = threadIdx.x & 31;
    const int half = lane >> 4, l16 = lane & 15;
    const int tile = blockIdx.x * 8 + wave;
    if (tile >= N_NODES / 16) return;           // wave-uniform: EXEC stays all-1s
    const int r0 = tile * 16;
    float* stage = s_stage[wave];

    // ---- h2 = H1tile[16x64] @ w2[64x32] ----
    for (int nt = 0; nt < 2; ++nt) {
        v8f acc = {};
        const float* a_base = H1 + (size_t)(r0 + l16) * HID + 2 * half;
        const float* b_base = s_w2 + (2 * half) * OUT_D + nt * 16 + l16;
        for (int k0 = 0; k0 < HID; k0 += 4) {
            v2f a, b;
            a.x = a_base[k0];
            a.y = a_base[k0 + 1];
            b.x = b_base[k0 * OUT_D];
            b.y = b_base[(k0 + 1) * OUT_D];
            acc = wmma_k4(a, b, acc);
        }
#pragma unroll
        for (int r = 0; r < 8; ++r) {
            int m = r + 8 * half, n = nt * 16 + l16;
            float v = acc[r];
            out_h2[(size_t)(r0 + m) * OUT_D + n] = v;
            stage[m * HID + n] = v;             // h2 in stage cols [0,32)
        }
    }

    // ---- h3 = elu(h2[16x32] @ l1w^T[32x64] + l1b) ----
    v8f acc3[4];
#pragma unroll
    for (int nt = 0; nt < 4; ++nt) {
        v8f acc = {};
        for (int k0 = 0; k0 < OUT_D; k0 += 4) {
            v2f a, b;
            a.x = stage[l16 * HID + k0 + 2 * half];
            a.y = stage[l16 * HID + k0 + 2 * half + 1];
            // B[k][n] = l1w[n*32 + k]
            b.x = s_l1w[(nt * 16 + l16) * OUT_D + k0 + 2 * half];
            b.y = s_l1w[(nt * 16 + l16) * OUT_D + k0 + 2 * half + 1];
            acc = wmma_k4(a, b, acc);
        }
        acc3[nt] = acc;
    }
#pragma unroll
    for (int nt = 0; nt < 4; ++nt)
#pragma unroll
        for (int r = 0; r < 8; ++r) {
            int m = r + 8 * half, n = nt * 16 + l16;
            float v = acc3[nt][r] + s_l1b[n];
            v = v > 0.0f ? v : (expf(v) - 1.0f);
            stage[m * HID + n] = v;             // h3 overwrites stage [16x64]
        }

    // ---- h4 = h3[16x64] @ l2w^T[64x256] + l2b ----
    for (int nt = 0; nt < 16; ++nt) {
        v8f acc = {};
        const float* b_base = l2w + (size_t)(nt * 16 + l16) * HID + 2 * half;
        for (int k0 = 0; k0 < HID; k0 += 4) {
            v2f a, b;
            a.x = stage[l16 * HID + k0 + 2 * half];
            a.y = stage[l16 * HID + k0 + 2 * half + 1];
            // B[k][n] = l2w[n*64 + k]
            b.x = b_base[k0];
            b.y = b_base[k0 + 1];
            acc = wmma_k4(a, b, acc);
        }
#pragma unroll
        for (int r = 0; r < 8; ++r) {
            int m = r + 8 * half, n = nt * 16 + l16;
            out_h4[(size_t)(r0 + m) * IN_D + n] = acc[r] + s_l2b[n];
        }
    }
}

extern "C" void kernel_launch(void* const* d_in, const int* in_sizes, int n_in,
                              void* d_out, int out_size, void* d_ws, size_t ws_size,
                              hipStream_t stream) {
    const float* features = (const float*)d_in[0];
    const int*   edge     = (const int*)d_in[1];    // [2,E] int32
    const float* w1       = (const float*)d_in[2];
    const float* att_src  = (const float*)d_in[3];
    const float* att_dst  = (const float*)d_in[4];
    const float* w2       = (const float*)d_in[5];
    const float* l1w      = (const float*)d_in[6];
    const float* l1b      = (const float*)d_in[7];
    const float* l2w      = (const float*)d_in[8];
    const float* l2b      = (const float*)d_in[9];

    const int* src = edge;
    const int* dst = edge + E_EDGES;

    float* ws      = (float*)d_ws;
    float* H       = ws;                                  // N*HID
    float* alpha_s = H + (size_t)N_NODES * HID;           // N
    float* alpha_d = alpha_s + N_NODES;                   // N
    float* nmax    = alpha_d + N_NODES;                   // N
    float* nsum    = nmax + N_NODES;                      // N
    float* eval_   = nsum + N_NODES;                      // E
    float* h1agg   = eval_ + E_EDGES;                     // N*HID

    float* out_h2 = (float*)d_out;
    float* out_h4 = out_h2 + (size_t)N_NODES * OUT_D;

    const int ROW_TILES = N_NODES / 16;                   // 6250 (exact)

    k_gemm1<<<ROW_TILES / 2, 256, 0, stream>>>(features, w1, H);
    k_alpha<<<(N_NODES + 255) / 256, 256, 0, stream>>>(H, att_src, att_dst, alpha_s, alpha_d);
    k_init<<<(N_NODES * HID + 255) / 256, 256, 0, stream>>>(h1agg, nmax, nsum);
    k_edge1<<<(E_EDGES + 255) / 256, 256, 0, stream>>>(src, dst, alpha_s, alpha_d, eval_, nmax);
    k_edge2<<<(E_EDGES + 255) / 256, 256, 0, stream>>>(dst, eval_, nmax, nsum);
    {
        long long total = (long long)E_EDGES * 32;
        k_edge3<<<(unsigned int)((total + 255) / 256), 256, 0, stream>>>(src, dst, eval_, nsum, H, h1agg);
    }
    k_elu<<<(N_NODES * HID + 255) / 256, 256, 0, stream>>>(h1agg);
    k_post<<<(ROW_TILES + 7) / 8, 256, 0, stream>>>(h1agg, w2, l1w, l1b, l2w, l2b, out_h2, out_h4);
}